// PoolingFilter_72069551227094
// MI455X (gfx1250) — compile-verified
//
#include <hip/hip_runtime.h>

typedef __attribute__((ext_vector_type(2))) float v2f;
typedef __attribute__((ext_vector_type(8))) float v8f;

#define B_  8
#define N_  512
#define F_  256
#define K_  64
#define BETA (1.0f / 512.0f)

__global__ __launch_bounds__(256) void pooling_filter_moments_wmma(
    const float* __restrict__ data,   // (B, N, F)
    const float* __restrict__ sp,     // (K,)
    float* __restrict__ out)          // (B, F, K)
{
    __shared__ float part[16][16][4];   // [n_chunk][f_local][moment]
    __shared__ float Mom[4][16];        // [moment M1..M4][f_local]
    __shared__ float tile[16][K_];      // unnormalized 16f x 64k tile
    __shared__ float invnorm[16];

    const int tid = threadIdx.x;
    const int b   = blockIdx.x >> 4;          // 8 batches
    const int f0  = (blockIdx.x & 15) << 4;   // 16 f-tiles of 16
    const int fl  = tid & 15;
    const int nc  = tid >> 4;                 // 16 chunks of 32 n-rows

    // ---- Phase A: streaming moment partials (bandwidth-bound pass) ----
    const float* p = data + ((size_t)(b * N_ + nc * 32) * F_) + (f0 + fl);
    float s1 = 0.f, s2 = 0.f, s3 = 0.f, s4 = 0.f;
    #pragma unroll 8
    for (int i = 0; i < 32; ++i) {
        float x  = p[(size_t)i * F_];
        float x2 = x * x;
        s1 += x; s2 += x2; s3 += x2 * x; s4 += x2 * x2;
    }
    part[nc][fl][0] = s1;
    part[nc][fl][1] = s2;
    part[nc][fl][2] = s3;
    part[nc][fl][3] = s4;
    __syncthreads();

    // ---- Phase B: reduce n-chunk partials -> moments M1..M4 ----
    if (tid < 64) {
        const int m = tid >> 4, f = tid & 15;
        float acc = 0.f;
        #pragma unroll
        for (int c = 0; c < 16; ++c) acc += part[c][f][m];
        Mom[m][f] = acc;
    }
    __syncthreads();

    // ---- Phase C: wave 0 does the rank-4 GEMM with V_WMMA_F32_16X16X4_F32 ----
    if (tid < 32) {                // wave-uniform: waves 1..7 branch over this
        const int lane = tid;
        const int half = lane >> 4;     // 0 -> A holds K=0,1 (M1,M2); 1 -> K=2,3 (M3,M4)
        const int l15  = lane & 15;

        // A-matrix 16x4 fp32 layout: lanes 0-15 rows M, v0=K0,v1=K1; lanes 16-31 v0=K2,v1=K3
        v2f a;
        a.x = Mom[half * 2 + 0][l15];
        a.y = Mom[half * 2 + 1][l15];

        const float b2h = BETA * BETA * 0.5f;   // beta^2 / 2
        #pragma unroll
        for (int kt = 0; kt < 4; ++kt) {
            const float s  = sp[kt * 16 + l15];
            const float ss = s * s;
            const float c0 = 1.0f + BETA * ss + b2h * ss * ss;

            // B-matrix 4x16: rows = coefficient index (c1..c4), cols = bins
            v2f bm;
            if (half == 0) {
                bm.x = -2.0f * BETA * s - 4.0f * b2h * ss * s;  // c1
                bm.y =  BETA + 6.0f * b2h * ss;                 // c2
            } else {
                bm.x = -4.0f * b2h * s;                         // c3
                bm.y =  b2h;                                    // c4
            }

            // C seeded with constant term N * c0(s_k): per-lane column value
            const float cval = (float)N_ * c0;
            v8f c;
            #pragma unroll
            for (int i = 0; i < 8; ++i) c[i] = cval;

            v8f d = __builtin_amdgcn_wmma_f32_16x16x4_f32(
                /*neg_a=*/false, a, /*neg_b=*/false, bm,
                /*c_mod=*/(short)0, c, /*reuse_a=*/false, /*reuse_b=*/false);

            // D layout: VGPR i -> row (i + 8*half), col = lane&15
            #pragma unroll
            for (int i = 0; i < 8; ++i) {
                tile[i + half * 8][kt * 16 + l15] = d[i];
            }
        }
    }
    __syncthreads();

    // ---- Phase D: per-row normalization coefficient ----
    if (tid < 16) {
        float acc = 0.f;
        #pragma unroll
        for (int k = 0; k < K_; ++k) acc += tile[tid][k];
        invnorm[tid] = 1.0f / acc;
    }
    __syncthreads();

    // ---- Phase E: write normalized tile (coalesced) ----
    #pragma unroll
    for (int i = 0; i < 4; ++i) {
        const int idx = tid * 4 + i;          // 0..1023
        const int row = idx >> 6;
        const int col = idx & 63;
        out[((size_t)(b * F_ + f0 + row) * K_) + col] = tile[row][col] * invnorm[row];
    }
}

extern "C" void kernel_launch(void* const* d_in, const int* in_sizes, int n_in,
                              void* d_out, int out_size, void* d_ws, size_t ws_size,
                              hipStream_t stream) {
    (void)in_sizes; (void)n_in; (void)out_size; (void)d_ws; (void)ws_size;
    const float* data = (const float*)d_in[0];   // (8, 512, 256) fp32
    const float* sp   = (const float*)d_in[1];   // (64,) fp32
    float* out        = (float*)d_out;           // (8, 256, 64) fp32

    dim3 grid(B_ * (F_ / 16));   // 128 workgroups
    dim3 block(256);             // 8 wave32 waves
    pooling_filter_moments_wmma<<<grid, block, 0, stream>>>(data, sp, out);
}